// MoE_61718680044029
// MI455X (gfx1250) — compile-verified
//
#include <hip/hip_runtime.h>
#include <math.h>

// MoE (DeepSeek-style group-limited routing) for MI455X / gfx1250.
// fp32 end-to-end using V_WMMA_F32_16X16X4_F32 (exact-precision tensor op).
// TILE_M=32: two M-subtiles per wave share each staged weight tile, halving
// L2 weight traffic and reaching a 1:1 ds-load:wmma ratio in phase 1.
//
// Workspace layout (requires ws_size >= ~64.6 MB for the default shapes):
//   counts[16] | tlist[E*T] | wsel[T*4] | scores[T*16] | yslot[T*4*D]

typedef __attribute__((ext_vector_type(2))) float v2f;
typedef __attribute__((ext_vector_type(4))) float v4f;
typedef __attribute__((ext_vector_type(8))) float v8f;

#define DDIM 1024
#define IDIM 1024
#define NEXP 16
#define KSEL 4
#define NGRP 4
#define ROUTE_SCALE 2.5f
#define TILE_M 32
#define KBLK 32          // staged K-block for weight tiles
#define XS_STRIDE 1028   // 1024+4 pad: LDS bank = (4m + k) & 63 -> conflict-free
#define WS_STRIDE 36     // 32+4 pad:   LDS bank = (36n + k) & 63 -> conflict-free
#define NWAVE 8

// ---------- kernel 0: zero expert counters (re-run on every graph replay) ----------
__global__ void moe_zero_counts(int* counts) {
  if (threadIdx.x < NEXP) counts[threadIdx.x] = 0;
}

// ---------- kernel 1: router scores  s = sigmoid(x @ Wg^T) ----------
__global__ void moe_gate_scores(const float* __restrict__ x,
                                const float* __restrict__ Wg,
                                float* __restrict__ scores, int T) {
  const int e  = threadIdx.x & (NEXP - 1);
  const int tl = threadIdx.x >> 4;
  const int t  = blockIdx.x * 16 + tl;
  if (t >= T) return;
  const float* xr = x  + (size_t)t * DDIM;
  const float* wr = Wg + (size_t)e * DDIM;
  float acc = 0.f;
  for (int d = 0; d < DDIM; d += 4) {
    v4f xv = *(const v4f*)(xr + d);
    v4f wv = *(const v4f*)(wr + d);
    acc += xv.x * wv.x + xv.y * wv.y + xv.z * wv.z + xv.w * wv.w;
  }
  scores[(size_t)t * NEXP + e] = 1.0f / (1.0f + __expf(-acc));
}

// ---------- kernel 2: group-limited top-k routing + expert token lists ----------
__global__ void moe_route(const float* __restrict__ scores,
                          float* __restrict__ wsel,
                          int* __restrict__ tlist,
                          int* __restrict__ counts, int T) {
  const int t = blockIdx.x * blockDim.x + threadIdx.x;
  if (t >= T) return;

  float s[NEXP];
#pragma unroll
  for (int e = 0; e < NEXP; ++e) s[e] = scores[(size_t)t * NEXP + e];

  // group scores = amax within each group of 4 experts
  float gs[NGRP];
#pragma unroll
  for (int g = 0; g < NGRP; ++g) {
    float m = s[4 * g];
    m = fmaxf(m, s[4 * g + 1]);
    m = fmaxf(m, s[4 * g + 2]);
    m = fmaxf(m, s[4 * g + 3]);
    gs[g] = m;
  }
  // top-2 groups (ties -> lowest index, matching lax.top_k)
  int g0 = 0;
#pragma unroll
  for (int g = 1; g < NGRP; ++g) if (gs[g] > gs[g0]) g0 = g;
  int g1 = (g0 == 0) ? 1 : 0;
#pragma unroll
  for (int g = 0; g < NGRP; ++g)
    if (g != g0 && g != g1 && gs[g] > gs[g1]) g1 = g;

  // mask non-selected groups, then top-4 experts
  float m2[NEXP];
#pragma unroll
  for (int e = 0; e < NEXP; ++e) {
    const int g = e >> 2;
    m2[e] = (g == g0 || g == g1) ? s[e] : -1.0f;  // sigmoid outputs > 0
  }
  int   sel[KSEL];
  float sw[KSEL];
  float wsum = 0.f;
#pragma unroll
  for (int k = 0; k < KSEL; ++k) {
    int best = 0; float bv = -2.0f;
#pragma unroll
    for (int e = 0; e < NEXP; ++e)
      if (m2[e] > bv) { bv = m2[e]; best = e; }
    sel[k] = best; sw[k] = s[best]; wsum += s[best];
    m2[best] = -2.0f;
  }
  const float inv = ROUTE_SCALE / wsum;
#pragma unroll
  for (int k = 0; k < KSEL; ++k) {
    wsel[(size_t)t * KSEL + k] = sw[k] * inv;
    const int pos = atomicAdd(&counts[sel[k]], 1);
    tlist[(size_t)sel[k] * T + pos] = (t << 2) | k;  // encode (token, k-slot)
  }
}

// ---------- kernel 3: fused expert FFN via fp32 WMMA ----------
// One workgroup = (expert, 32-token tile). Phase 1: H = silu(X@W1^T)*(X@W3^T)
// kept in LDS. Phase 2: Y = H@W2^T + b2, rows scattered to per-slot buffer.
__global__ void moe_expert_ffn(const float* __restrict__ x,
                               const float* __restrict__ W1,
                               const float* __restrict__ b1,
                               const float* __restrict__ W2,
                               const float* __restrict__ b2,
                               const float* __restrict__ W3,
                               const float* __restrict__ b3,
                               const int* __restrict__ tlist,
                               const int* __restrict__ counts,
                               float* __restrict__ yslot, int T) {
  extern __shared__ float smem[];
  float* Xs  = smem;                                     // 32 x XS_STRIDE
  float* Hs  = smem + TILE_M * XS_STRIDE;                // 32 x XS_STRIDE
  float* Wst = Hs + TILE_M * XS_STRIDE;                  // NWAVE x 2 x 16 x WS_STRIDE
  int* slotLds = (int*)(Wst + NWAVE * 2 * 16 * WS_STRIDE);

  const int maxTiles = T / TILE_M;
  const int e    = blockIdx.x / maxTiles;
  const int tile = blockIdx.x % maxTiles;
  const int cnt  = counts[e];
  const int base = tile * TILE_M;
  if (base >= cnt) return;                 // uniform across the block
  const int rows = min(TILE_M, cnt - base);

  const int tid = threadIdx.x;
  if (tid < TILE_M)
    slotLds[tid] = (tid < rows) ? tlist[(size_t)e * T + base + tid] : -1;
  __syncthreads();

  // stage X tile into LDS (zero-padded rows so WMMA sees clean data)
  for (int i = tid; i < TILE_M * (DDIM / 4); i += 256) {
    const int m  = i >> 8;                 // DDIM/4 == 256 float4 per row
    const int c4 = i & 255;
    v4f val = {0.f, 0.f, 0.f, 0.f};
    const int slot = slotLds[m];
    if (slot >= 0)
      val = *(const v4f*)(x + (size_t)(slot >> 2) * DDIM + c4 * 4);
    *(v4f*)(Xs + m * XS_STRIDE + c4 * 4) = val;
  }
  __syncthreads();

  const int w    = tid >> 5;
  const int lane = tid & 31;
  const int nidx = lane & 15;            // A-row M / B-col N index
  const int koff = (lane >> 4) << 1;     // K sub-offset for upper half-wave
  const int mb   = (lane >> 4) << 3;     // C/D row base (0 or 8)
  float* Wa = Wst + w * (2 * 16 * WS_STRIDE);
  float* Wb = Wa + 16 * WS_STRIDE;

  const float* W1e = W1 + (size_t)e * IDIM * DDIM;
  const float* W3e = W3 + (size_t)e * IDIM * DDIM;

  // ---- phase 1 ----
  for (int nt = w; nt < IDIM / 16; nt += NWAVE) {
    const int i0 = nt * 16;
    v8f c1a = {0,0,0,0,0,0,0,0}, c1b = {0,0,0,0,0,0,0,0};
    v8f c3a = {0,0,0,0,0,0,0,0}, c3b = {0,0,0,0,0,0,0,0};
    for (int kb = 0; kb < DDIM; kb += KBLK) {
      // stage W1/W3 [16 x KBLK] sub-tiles (coalesced) into padded LDS
#pragma unroll
      for (int j = lane; j < 16 * (KBLK / 4); j += 32) {
        const int n = j >> 3, c4 = j & 7;       // KBLK/4 == 8 float4 per row
        *(v4f*)(Wa + n * WS_STRIDE + c4 * 4) =
            *(const v4f*)(W1e + (size_t)(i0 + n) * DDIM + kb + c4 * 4);
        *(v4f*)(Wb + n * WS_STRIDE + c4 * 4) =
            *(const v4f*)(W3e + (size_t)(i0 + n) * DDIM + kb + c4 * 4);
      }
      if (kb + KBLK < DDIM) {
        __builtin_prefetch(W1e + (size_t)(i0 + nidx) * DDIM + kb + KBLK, 0, 1);
        __builtin_prefetch(W3e + (size_t)(i0 + nidx) * DDIM + kb + KBLK, 0, 1);
      }
      asm volatile("s_wait_dscnt 0" ::: "memory");
#pragma unroll
      for (int kk = 0; kk < KBLK; kk += 4) {
        v2f a0, a1, bv1, bv3;
        const float* ap = Xs + nidx * XS_STRIDE + kb + kk + koff;
        a0.x = ap[0];                 a0.y = ap[1];
        a1.x = ap[16 * XS_STRIDE];    a1.y = ap[16 * XS_STRIDE + 1];
        const float* bp1 = Wa + nidx * WS_STRIDE + kk + koff;
        bv1.x = bp1[0]; bv1.y = bp1[1];
        const float* bp3 = Wb + nidx * WS_STRIDE + kk + koff;
        bv3.x = bp3[0]; bv3.y = bp3[1];
        c1a = __builtin_amdgcn_wmma_f32_16x16x4_f32(false, a0, false, bv1,
                                                    (short)0, c1a, false, false);
        c1b = __builtin_amdgcn_wmma_f32_16x16x4_f32(false, a1, false, bv1,
                                                    (short)0, c1b, false, false);
        c3a = __builtin_amdgcn_wmma_f32_16x16x4_f32(false, a0, false, bv3,
                                                    (short)0, c3a, false, false);
        c3b = __builtin_amdgcn_wmma_f32_16x16x4_f32(false, a1, false, bv3,
                                                    (short)0, c3b, false, false);
      }
    }
    const float bias1 = b1[(size_t)e * IDIM + i0 + nidx];
    const float bias3 = b3[(size_t)e * IDIM + i0 + nidx];
#pragma unroll
    for (int r = 0; r < 8; ++r) {
      const float u0 = c1a[r] + bias1;
      const float g0 = c3a[r] + bias3;
      Hs[(mb + r) * XS_STRIDE + i0 + nidx] =
          (u0 / (1.0f + __expf(-u0))) * g0;            // silu(u) * g
      const float u1 = c1b[r] + bias1;
      const float g1v = c3b[r] + bias3;
      Hs[(mb + r + 16) * XS_STRIDE + i0 + nidx] =
          (u1 / (1.0f + __expf(-u1))) * g1v;
    }
  }
  __syncthreads();

  // ---- phase 2 ----
  const float* W2e = W2 + (size_t)e * DDIM * IDIM;
  for (int nt = w; nt < DDIM / 16; nt += NWAVE) {
    const int d0 = nt * 16;
    v8f ca = {0,0,0,0,0,0,0,0}, cb = {0,0,0,0,0,0,0,0};
    for (int kb = 0; kb < IDIM; kb += KBLK) {
#pragma unroll
      for (int j = lane; j < 16 * (KBLK / 4); j += 32) {
        const int n = j >> 3, c4 = j & 7;
        *(v4f*)(Wa + n * WS_STRIDE + c4 * 4) =
            *(const v4f*)(W2e + (size_t)(d0 + n) * IDIM + kb + c4 * 4);
      }
      if (kb + KBLK < IDIM)
        __builtin_prefetch(W2e + (size_t)(d0 + nidx) * IDIM + kb + KBLK, 0, 1);
      asm volatile("s_wait_dscnt 0" ::: "memory");
#pragma unroll
      for (int kk = 0; kk < KBLK; kk += 4) {
        v2f a0, a1, bv;
        const float* ap = Hs + nidx * XS_STRIDE + kb + kk + koff;
        a0.x = ap[0];                 a0.y = ap[1];
        a1.x = ap[16 * XS_STRIDE];    a1.y = ap[16 * XS_STRIDE + 1];
        const float* bp = Wa + nidx * WS_STRIDE + kk + koff;
        bv.x = bp[0]; bv.y = bp[1];
        ca = __builtin_amdgcn_wmma_f32_16x16x4_f32(false, a0, false, bv,
                                                   (short)0, ca, false, false);
        cb = __builtin_amdgcn_wmma_f32_16x16x4_f32(false, a1, false, bv,
                                                   (short)0, cb, false, false);
      }
    }
    const float bias2 = b2[(size_t)e * DDIM + d0 + nidx];
#pragma unroll
    for (int r = 0; r < 8; ++r) {
      const int slot0 = slotLds[mb + r];
      if (slot0 >= 0)
        yslot[(size_t)slot0 * DDIM + d0 + nidx] = ca[r] + bias2;
      const int slot1 = slotLds[mb + r + 16];
      if (slot1 >= 0)
        yslot[(size_t)slot1 * DDIM + d0 + nidx] = cb[r] + bias2;
    }
  }
}

// ---------- kernel 4: weighted combine of the 4 slots per token ----------
__global__ void moe_combine(const float* __restrict__ yslot,
                            const float* __restrict__ wsel,
                            float* __restrict__ out, int T) {
  const int idx = blockIdx.x * blockDim.x + threadIdx.x;  // float4 granule
  if (idx >= T * (DDIM / 4)) return;
  const int t  = idx >> 8;
  const int c4 = idx & 255;
  v4f acc = {0.f, 0.f, 0.f, 0.f};
#pragma unroll
  for (int k = 0; k < KSEL; ++k) {
    const float wk = wsel[(size_t)t * KSEL + k];
    const v4f yv = *(const v4f*)(yslot + (size_t)(t * KSEL + k) * DDIM + c4 * 4);
    acc.x += wk * yv.x; acc.y += wk * yv.y;
    acc.z += wk * yv.z; acc.w += wk * yv.w;
  }
  *(v4f*)(out + (size_t)t * DDIM + c4 * 4) = acc;
}

extern "C" void kernel_launch(void* const* d_in, const int* in_sizes, int n_in,
                              void* d_out, int out_size, void* d_ws, size_t ws_size,
                              hipStream_t stream) {
  const float* x  = (const float*)d_in[0];
  const float* Wg = (const float*)d_in[1];
  const float* W1 = (const float*)d_in[2];
  const float* b1 = (const float*)d_in[3];
  const float* W2 = (const float*)d_in[4];
  const float* b2 = (const float*)d_in[5];
  const float* W3 = (const float*)d_in[6];
  const float* b3 = (const float*)d_in[7];
  float* out = (float*)d_out;
  const int T = in_sizes[0] / DDIM;   // 4096 for default shapes
  (void)n_in; (void)out_size; (void)ws_size;

  char* ws = (char*)d_ws;
  size_t off = 0;
  int* counts   = (int*)(ws + off);   off += 256;
  int* tlist    = (int*)(ws + off);   off += (size_t)NEXP * T * sizeof(int);
  float* wsel   = (float*)(ws + off); off += (size_t)KSEL * T * sizeof(float);
  float* scores = (float*)(ws + off); off += (size_t)NEXP * T * sizeof(float);
  off = (off + 255) & ~(size_t)255;
  float* yslot  = (float*)(ws + off); // T*KSEL*DDIM floats (~64 MB)

  moe_zero_counts<<<dim3(1), dim3(32), 0, stream>>>(counts);
  moe_gate_scores<<<dim3((T + 15) / 16), dim3(256), 0, stream>>>(x, Wg, scores, T);
  moe_route<<<dim3((T + 255) / 256), dim3(256), 0, stream>>>(scores, wsel, tlist, counts, T);

  const int maxTiles = T / TILE_M;
  const size_t lds =
      ((size_t)(2 * TILE_M * XS_STRIDE + NWAVE * 2 * 16 * WS_STRIDE)) * sizeof(float)
      + TILE_M * sizeof(int);  // ~293 KB (<= 320 KB/WGP on CDNA5)
  moe_expert_ffn<<<dim3(NEXP * maxTiles), dim3(256), lds, stream>>>(
      x, W1, b1, W2, b2, W3, b3, tlist, counts, yslot, T);

  moe_combine<<<dim3((T * (DDIM / 4) + 255) / 256), dim3(256), 0, stream>>>(
      yslot, wsel, out, T);
}